// FeaStNet_33328946217321
// MI455X (gfx1250) — compile-verified
//
#include <hip/hip_runtime.h>
#include <math.h>

#define NNODES 10000
#define NHEADS 8

typedef __attribute__((ext_vector_type(2))) float v2f;
typedef __attribute__((ext_vector_type(8))) float v8f;

// ============================================================================
// WMMA f32 GEMM:  C[M,Nc] = A[M,K] @ B[K,Nc]  (+ optional bias, optional relu)
// Register-blocked + software-pipelined: one wave computes a 16x(16*CT) slab
// with CT accumulators; the next K-step's A/B fragments are loaded into
// separate registers while the current CT wmmas execute, so the
// s_wait_loadcnt before each wmma group covers loads issued a full
// iteration earlier (latency hidden behind matrix work).
// Requires: M%16==0, K%4==0 (>=8), Nc % (16*CT*blockDim.y) == 0. No divergence.
// Fragment layouts per CDNA5 ISA 7.12.2:
//   A 16x4 : lanes 0-15 hold {K=0,K=1}, lanes 16-31 hold {K=2,K=3}, lane%16 = M
//   B 4x16 : lanes 0-15 hold {K=0,K=1}, lanes 16-31 hold {K=2,K=3}, lane%16 = N
//   C 16x16: VGPR v -> M = v + 8*(lane/16), N = lane%16
// ============================================================================
template <int CT>
__global__ void gemm_wmma_f32(const float* __restrict__ A, const float* __restrict__ B,
                              float* __restrict__ C, const float* __restrict__ bias,
                              int M, int K, int Nc, int relu)
{
    const int lane = threadIdx.x;      // 0..31
    const int lr   = lane & 15;
    const int half = lane >> 4;        // 0 or 1
    const int row0 = blockIdx.y * 16;
    const int col0 = (blockIdx.x * blockDim.y + threadIdx.y) * (16 * CT);

    v8f acc[CT];
#pragma unroll
    for (int j = 0; j < CT; ++j) acc[j] = (v8f){};

    const float* __restrict__ Arow = A + (size_t)(row0 + lr) * K + 2 * half;
    const float* __restrict__ B0   = B + (size_t)(2 * half) * Nc + col0 + lr;
    const float* __restrict__ B1   = B0 + Nc;
    const size_t bstep = (size_t)4 * Nc;

    // ---- prologue: load fragments for k = 0 ----
    v2f a_cur, b_cur[CT];
    {
        float2 av = *(const float2*)Arow;
        a_cur.x = av.x; a_cur.y = av.y;
#pragma unroll
        for (int j = 0; j < CT; ++j) {
            b_cur[j].x = B0[j * 16];
            b_cur[j].y = B1[j * 16];
        }
    }

    // ---- steady state: prefetch k+4 while computing k ----
    for (int k = 0; k < K - 4; k += 4) {
        Arow += 4;
        B0   += bstep;
        B1   += bstep;
        v2f a_nxt, b_nxt[CT];
        {
            float2 av = *(const float2*)Arow;
            a_nxt.x = av.x; a_nxt.y = av.y;
#pragma unroll
            for (int j = 0; j < CT; ++j) {
                b_nxt[j].x = B0[j * 16];
                b_nxt[j].y = B1[j * 16];
            }
        }
#pragma unroll
        for (int j = 0; j < CT; ++j) {
            acc[j] = __builtin_amdgcn_wmma_f32_16x16x4_f32(false, a_cur, false, b_cur[j],
                                                           (short)0, acc[j], false, false);
        }
        a_cur = a_nxt;
#pragma unroll
        for (int j = 0; j < CT; ++j) b_cur[j] = b_nxt[j];
    }

    // ---- epilogue: final K-step ----
#pragma unroll
    for (int j = 0; j < CT; ++j) {
        acc[j] = __builtin_amdgcn_wmma_f32_16x16x4_f32(false, a_cur, false, b_cur[j],
                                                       (short)0, acc[j], false, false);
    }

#pragma unroll
    for (int j = 0; j < CT; ++j) {
        const int col = col0 + j * 16 + lr;
        float bv = bias ? bias[col] : 0.0f;
#pragma unroll
        for (int v = 0; v < 8; ++v) {
            float out = acc[j][v] + bv;
            if (relu) out = fmaxf(out, 0.0f);
            C[(size_t)(row0 + 8 * half + v) * Nc + col] = out;
        }
    }
}

// ============================================================================
// Column-wise batchnorm statistics (training mode, biased variance).
// One block per channel; LDS tree reduction over N rows.
// ============================================================================
__global__ void bn_stats_kernel(const float* __restrict__ h, float* __restrict__ mu,
                                float* __restrict__ rsig, int N, int C)
{
    __shared__ float ssum[256];
    __shared__ float ssq[256];
    const int c = blockIdx.x;
    float s = 0.f, sq = 0.f;
    for (int n = threadIdx.x; n < N; n += blockDim.x) {
        float v = h[(size_t)n * C + c];
        s += v; sq += v * v;
    }
    ssum[threadIdx.x] = s; ssq[threadIdx.x] = sq;
    __syncthreads();
    for (int off = 128; off > 0; off >>= 1) {
        if ((int)threadIdx.x < off) {
            ssum[threadIdx.x] += ssum[threadIdx.x + off];
            ssq[threadIdx.x]  += ssq[threadIdx.x + off];
        }
        __syncthreads();
    }
    if (threadIdx.x == 0) {
        float m   = ssum[0] / (float)N;
        float var = ssq[0] / (float)N - m * m;
        mu[c]   = m;
        rsig[c] = rsqrtf(fmaxf(var, 0.f) + 1e-5f);
    }
}

__global__ void bn_apply_relu_kernel(const float* __restrict__ hin, float* __restrict__ hout,
                                     const float* __restrict__ mu, const float* __restrict__ rsig,
                                     const float* __restrict__ gamma, const float* __restrict__ beta,
                                     int N, int C)
{
    int idx = blockIdx.x * blockDim.x + threadIdx.x;
    if (idx >= N * C) return;
    int c = idx % C;
    float v = (hin[idx] - mu[c]) * rsig[c] * gamma[c] + beta[c];
    hout[idx] = fmaxf(v, 0.f);
}

// ============================================================================
// Input stage: bn(pos) || x  -> lin0 (4->16) -> relu
// ============================================================================
__global__ void input_lin0_kernel(const float* __restrict__ pos, const float* __restrict__ x,
                                  const float* __restrict__ mu, const float* __restrict__ rsig,
                                  const float* __restrict__ g0, const float* __restrict__ b0,
                                  const float* __restrict__ W, const float* __restrict__ b,
                                  float* __restrict__ h, int N)
{
    int n = blockIdx.x * blockDim.x + threadIdx.x;
    if (n >= N) return;
    float in4[4];
    in4[0] = (pos[n * 2 + 0] - mu[0]) * rsig[0] * g0[0] + b0[0];
    in4[1] = (pos[n * 2 + 1] - mu[1]) * rsig[1] * g0[1] + b0[1];
    in4[2] = x[n * 2 + 0];
    in4[3] = x[n * 2 + 1];
#pragma unroll
    for (int j = 0; j < 16; ++j) {
        float acc = b[j];
#pragma unroll
        for (int i = 0; i < 4; ++i) acc += in4[i] * W[i * 16 + j];
        h[n * 16 + j] = fmaxf(acc, 0.f);
    }
}

// ============================================================================
// hu = h @ u   ([N,K] x [K,8] -> [N,8])
// ============================================================================
__global__ void hu_kernel(const float* __restrict__ h, const float* __restrict__ u,
                          float* __restrict__ hu, int N, int K)
{
    int n = blockIdx.x * blockDim.x + threadIdx.x;
    if (n >= N) return;
    float acc[NHEADS];
#pragma unroll
    for (int t = 0; t < NHEADS; ++t) acc[t] = 0.f;
    for (int k = 0; k < K; ++k) {
        float hv = h[(size_t)n * K + k];
#pragma unroll
        for (int t = 0; t < NHEADS; ++t) acc[t] += hv * u[k * NHEADS + t];
    }
#pragma unroll
    for (int t = 0; t < NHEADS; ++t) hu[n * NHEADS + t] = acc[t];
}

// ============================================================================
// Per-edge head softmax: q[e] = softmax(hu[src]-hu[dst]+c). Edges e>=E are self loops.
// ============================================================================
__global__ void edge_q_kernel(const float* __restrict__ hu, const int* __restrict__ src,
                              const int* __restrict__ dst, const float* __restrict__ cbias,
                              float* __restrict__ q, int E, int N)
{
    int e = blockIdx.x * blockDim.x + threadIdx.x;
    if (e >= E + N) return;
    int s, d;
    if (e < E) { s = src[e]; d = dst[e]; } else { s = e - E; d = s; }
    float z[NHEADS];
    float m = -1e30f;
#pragma unroll
    for (int t = 0; t < NHEADS; ++t) {
        z[t] = hu[s * NHEADS + t] - hu[d * NHEADS + t] + cbias[t];
        m = fmaxf(m, z[t]);
    }
    float sum = 0.f;
#pragma unroll
    for (int t = 0; t < NHEADS; ++t) { z[t] = expf(z[t] - m); sum += z[t]; }
    float inv = 1.f / sum;
#pragma unroll
    for (int t = 0; t < NHEADS; ++t) q[(size_t)e * NHEADS + t] = z[t] * inv;
}

// ============================================================================
// Edge aggregation: agg[dst] += sum_h q[e,h] * hW[src, h, c4:c4+4]
// One thread per (edge, 4-channel chunk); gathers stay L2-resident (hW <= 82MB).
// ============================================================================
__global__ void edge_agg_kernel(const float* __restrict__ q, const float* __restrict__ hW,
                                const int* __restrict__ src, const int* __restrict__ dst,
                                float* __restrict__ agg, int E, int N, int Co)
{
    const int chunks = Co >> 2;
    long long idx = (long long)blockIdx.x * blockDim.x + threadIdx.x;
    long long total = (long long)(E + N) * chunks;
    if (idx >= total) return;
    int e  = (int)(idx / chunks);
    int c4 = ((int)(idx % chunks)) << 2;
    int s, d;
    if (e < E) { s = src[e]; d = dst[e]; } else { s = e - E; d = s; }
    const float* __restrict__ qe = q + (size_t)e * NHEADS;
    float4 acc = {0.f, 0.f, 0.f, 0.f};
#pragma unroll
    for (int t = 0; t < NHEADS; ++t) {
        float qv = qe[t];
        const float4 v = *(const float4*)(hW + ((size_t)s * NHEADS + t) * Co + c4);
        acc.x += qv * v.x; acc.y += qv * v.y; acc.z += qv * v.z; acc.w += qv * v.w;
    }
    float* out = agg + (size_t)d * Co + c4;
    atomicAdd(out + 0, acc.x);
    atomicAdd(out + 1, acc.y);
    atomicAdd(out + 2, acc.z);
    atomicAdd(out + 3, acc.w);
}

// ============================================================================
// Misc small kernels
// ============================================================================
__global__ void zero_kernel(float* __restrict__ p, int n)
{
    int i = blockIdx.x * blockDim.x + threadIdx.x;
    if (i < n) p[i] = 0.f;
}

__global__ void deg_kernel(float* __restrict__ deg, const int* __restrict__ dst, int E)
{
    int e = blockIdx.x * blockDim.x + threadIdx.x;
    if (e >= E) return;
    atomicAdd(&deg[dst[e]], 1.0f);
}

__global__ void rdeg_kernel(float* __restrict__ deg, int N)
{
    int n = blockIdx.x * blockDim.x + threadIdx.x;
    if (n >= N) return;
    deg[n] = 1.0f / (deg[n] + 1.0f);   // +1 for self loop; always >= 1
}

__global__ void finalize_kernel(float* __restrict__ agg, const float* __restrict__ rdeg,
                                const float* __restrict__ bias, int N, int C)
{
    int idx = blockIdx.x * blockDim.x + threadIdx.x;
    if (idx >= N * C) return;
    int n = idx / C, c = idx % C;
    agg[idx] = agg[idx] * rdeg[n] + bias[c];
}

__global__ void lin2_kernel(const float* __restrict__ h, const float* __restrict__ W,
                            const float* __restrict__ b, float* __restrict__ out, int N)
{
    int n = blockIdx.x * blockDim.x + threadIdx.x;
    if (n >= N) return;
    float a0 = b[0], a1 = b[1];
#pragma unroll 8
    for (int k = 0; k < 64; ++k) {
        float hv = h[n * 64 + k];
        a0 += hv * W[k * 2 + 0];
        a1 += hv * W[k * 2 + 1];
    }
    out[n * 2 + 0] = a0;
    out[n * 2 + 1] = a1;
}

// ============================================================================
// Host orchestration
// ============================================================================
extern "C" void kernel_launch(void* const* d_in, const int* in_sizes, int n_in,
                              void* d_out, int out_size, void* d_ws, size_t ws_size,
                              hipStream_t stream)
{
    (void)n_in; (void)out_size; (void)ws_size;
    const int N = NNODES;
    const float* pos = (const float*)d_in[0];
    const float* x   = (const float*)d_in[1];
    const int*   ei  = (const int*)d_in[2];
    const int E   = in_sizes[2] / 2;
    const int* src = ei;
    const int* dst = ei + E;
    const int ET  = E + N;

    // params, recursive dict-insertion order
    int pi = 3;
    const float* norm0_g = (const float*)d_in[pi++];
    const float* norm0_b = (const float*)d_in[pi++];
    const float* lin0_W  = (const float*)d_in[pi++];
    const float* lin0_b  = (const float*)d_in[pi++];
    const int Ci[6] = {16, 32, 64, 128, 256, 128};
    const int Co[6] = {32, 64, 128, 256, 128, 128};
    const float *cW[6], *cu[6], *cc[6], *cb[6], *ng[6], *nb[6];
    for (int i = 0; i < 6; ++i) {
        cW[i] = (const float*)d_in[pi++];
        cu[i] = (const float*)d_in[pi++];
        cc[i] = (const float*)d_in[pi++];
        cb[i] = (const float*)d_in[pi++];
        ng[i] = (const float*)d_in[pi++];
        nb[i] = (const float*)d_in[pi++];
    }
    const float* lin1_W = (const float*)d_in[pi++];
    const float* lin1_b = (const float*)d_in[pi++];
    const float* lin2_W = (const float*)d_in[pi++];
    const float* lin2_b = (const float*)d_in[pi++];

    // workspace layout (floats)
    float* ws   = (float*)d_ws;
    float* h    = ws;  ws += (size_t)N * 256;     // current features (max C=256)
    float* hpre = ws;  ws += (size_t)N * 256;     // agg / pre-BN buffer
    float* hW   = ws;  ws += (size_t)N * 2048;    // h @ W (max 8*256 cols)
    float* hub  = ws;  ws += (size_t)N * NHEADS;  // h @ u
    float* q    = ws;  ws += (size_t)ET * NHEADS; // edge softmax
    float* rdeg = ws;  ws += N;                   // 1/deg
    float* mu   = ws;  ws += 256;
    float* rsig = ws;  ws += 256;

    const int T = 256;

    // reciprocal degrees (shared by all layers; self loops included)
    zero_kernel<<<(N + T - 1) / T, T, 0, stream>>>(rdeg, N);
    deg_kernel<<<(E + T - 1) / T, T, 0, stream>>>(rdeg, dst, E);
    rdeg_kernel<<<(N + T - 1) / T, T, 0, stream>>>(rdeg, N);

    // input stage: batchnorm(pos) || x -> lin0 -> relu
    bn_stats_kernel<<<2, T, 0, stream>>>(pos, mu, rsig, N, 2);
    input_lin0_kernel<<<(N + T - 1) / T, T, 0, stream>>>(pos, x, mu, rsig, norm0_g, norm0_b,
                                                         lin0_W, lin0_b, h, N);

    // 6 FeaStConv layers
    for (int i = 0; i < 6; ++i) {
        const int K  = Ci[i];
        const int C  = Co[i];
        const int Nc = NHEADS * C;

        // hW = h @ W : wave slab = 16x64, block = 16x256  (Nc is 256..2048)
        dim3 bb(32, 4);
        dim3 gb(Nc / 256, N / 16);
        gemm_wmma_f32<4><<<gb, bb, 0, stream>>>(h, cW[i], hW, nullptr, N, K, Nc, 0);

        hu_kernel<<<(N + T - 1) / T, T, 0, stream>>>(h, cu[i], hub, N, K);
        edge_q_kernel<<<(ET + T - 1) / T, T, 0, stream>>>(hub, src, dst, cc[i], q, E, N);

        zero_kernel<<<(N * C + T - 1) / T, T, 0, stream>>>(hpre, N * C);
        long long total = (long long)ET * (C / 4);
        edge_agg_kernel<<<(unsigned)((total + T - 1) / T), T, 0, stream>>>(q, hW, src, dst,
                                                                           hpre, E, N, C);

        finalize_kernel<<<(N * C + T - 1) / T, T, 0, stream>>>(hpre, rdeg, cb[i], N, C);
        bn_stats_kernel<<<C, T, 0, stream>>>(hpre, mu, rsig, N, C);
        bn_apply_relu_kernel<<<(N * C + T - 1) / T, T, 0, stream>>>(hpre, h, mu, rsig,
                                                                    ng[i], nb[i], N, C);
    }

    // lin1: [N,128] @ [128,64] + b, relu (WMMA; one wave covers all 64 cols)
    {
        dim3 bb(32, 1);
        dim3 gb(1, N / 16);
        gemm_wmma_f32<4><<<gb, bb, 0, stream>>>(h, lin1_W, hpre, lin1_b, N, 128, 64, 1);
    }
    // lin2: [N,64] @ [64,2] + b
    lin2_kernel<<<(N + T - 1) / T, T, 0, stream>>>(hpre, lin2_W, lin2_b, (float*)d_out, N);
}